// NGCF_7464653161107
// MI455X (gfx1250) — compile-verified
//
#include <hip/hip_runtime.h>

typedef float v2f __attribute__((ext_vector_type(2)));
typedef float v8f __attribute__((ext_vector_type(8)));

#define LEAKY 0.2f
#define L2EPS 1e-12f

// ---------------- utility kernels ----------------

__global__ void fill_zero_f32(float* __restrict__ p, int n) {
  int i = blockIdx.x * blockDim.x + threadIdx.x;
  if (i < n) p[i] = 0.0f;
}

__global__ void deg_kernel(const int* __restrict__ ei, float* __restrict__ deg, int E) {
  int e = blockIdx.x * blockDim.x + threadIdx.x;
  if (e < E) atomicAdd(&deg[ei[e]], 1.0f);  // row indices = ei[0:E]
}

__global__ void dinv_kernel(float* __restrict__ d, int N) {
  int i = blockIdx.x * blockDim.x + threadIdx.x;
  if (i < N) {
    float v = d[i];
    d[i] = (v > 0.0f) ? rsqrtf(v) : 0.0f;
  }
}

__global__ void edgew_kernel(const int* __restrict__ ei, const float* __restrict__ dinv,
                             float* __restrict__ w, int E) {
  int e = blockIdx.x * blockDim.x + threadIdx.x;
  if (e < E) w[e] = dinv[ei[e]] * dinv[ei[E + e]];
}

// out[:, 0:64] = x   (out row stride = outStride floats)
__global__ void copy_x_kernel(const float* __restrict__ x, float* __restrict__ out,
                              int N, int outStride) {
  int t = blockIdx.x * blockDim.x + threadIdx.x;
  if (t < N * 16) {
    int r = t >> 4, q = t & 15;
    float4 v = reinterpret_cast<const float4*>(x)[(size_t)r * 16 + q];
    reinterpret_cast<float4*>(out + (size_t)r * outStride)[q] = v;
  }
}

// ---------------- sparse aggregation: ha[row] += w * h[col] ----------------
// 32 lanes per edge, 2 features per lane (float2 gather, two f32 atomics into L2)
__global__ void agg_kernel(const int* __restrict__ ei, const float* __restrict__ w,
                           const float* __restrict__ h, float* __restrict__ ha, int E) {
  long long t = (long long)blockIdx.x * blockDim.x + threadIdx.x;
  if (t >= (long long)E * 32) return;
  int e = (int)(t >> 5);
  int p = (int)(t & 31);
  int r = ei[e];
  int c = ei[E + e];
  float wv = w[e];
  v2f hv = *reinterpret_cast<const v2f*>(h + (size_t)c * 64 + p * 2);
  float* dst = ha + (size_t)r * 64 + p * 2;
  atomicAdd(dst + 0, wv * hv.x);
  atomicAdd(dst + 1, wv * hv.y);
}

// ---------------- dense layer: WMMA fp32 16x16x4 ----------------
// h_new = lrelu(ha@Wg + bg) + lrelu((h*ha)@Wi + bi)
// out[:, colOff:colOff+64] = l2norm(h_new) ; hnext = h_new (raw)
__launch_bounds__(256)
__global__ void layer_kernel(const float* __restrict__ h,  const float* __restrict__ ha,
                             const float* __restrict__ Wg, const float* __restrict__ bg,
                             const float* __restrict__ Wi, const float* __restrict__ bi,
                             float* __restrict__ hnext, float* __restrict__ out,
                             int N, int outStride, int colOff) {
  // LDS: weights stored pair-interleaved so a B fragment is one b64 read:
  // sW[ ((r>>1)*64 + c)*2 + (r&1) ] = W[r][c]  -> pair k2 at float2 index (k2*64 + c)
  __shared__ float sWg[64 * 64];
  __shared__ float sWi[64 * 64];
  __shared__ float sbg[64];
  __shared__ float sbi[64];

  int tid = threadIdx.x;
  for (int i = tid; i < 4096; i += 256) {
    int r = i >> 6, c = i & 63;
    int idx = (((r >> 1) * 64 + c) << 1) | (r & 1);
    sWg[idx] = Wg[i];
    sWi[idx] = Wi[i];
  }
  if (tid < 64) { sbg[tid] = bg[tid]; sbi[tid] = bi[tid]; }
  __syncthreads();

  int wave = tid >> 5, lane = tid & 31;
  int tiles = (N + 15) >> 4;
  int tile = blockIdx.x * 8 + wave;
  if (tile >= tiles) return;          // wave-uniform: EXEC stays all-ones for WMMA

  int m0 = tile << 4;
  int kh  = lane >> 4;                // lane half: carries K offset in A/B frags
  int l15 = lane & 15;

  int rowA = m0 + l15;
  int rowLd = rowA < N ? rowA : N - 1;          // clamp loads for partial tiles
  const float* haRow = ha + (size_t)rowLd * 64 + 2 * kh;
  const float* hRow  = h  + (size_t)rowLd * 64 + 2 * kh;

  v8f zero8 = {0.f, 0.f, 0.f, 0.f, 0.f, 0.f, 0.f, 0.f};
  v8f accg[4], acci[4];
#pragma unroll
  for (int nt = 0; nt < 4; nt++) { accg[nt] = zero8; acci[nt] = zero8; }

  const v2f* sWg2 = reinterpret_cast<const v2f*>(sWg);
  const v2f* sWi2 = reinterpret_cast<const v2f*>(sWi);

#pragma unroll
  for (int k = 0; k < 64; k += 4) {
    // A fragments (16x4 fp32): lane<16 -> K=k,k+1 ; lane>=16 -> K=k+2,k+3
    v2f ga = *reinterpret_cast<const v2f*>(haRow + k);
    v2f hv = *reinterpret_cast<const v2f*>(hRow + k);
    v2f ia = ga * hv;                          // (h * ha) fragment
    int k2 = (k >> 1) + kh;                    // pair row in LDS layout
#pragma unroll
    for (int nt = 0; nt < 4; nt++) {
      v2f bgf = sWg2[k2 * 64 + nt * 16 + l15]; // {W[kb][col], W[kb+1][col]}
      v2f bif = sWi2[k2 * 64 + nt * 16 + l15];
      accg[nt] = __builtin_amdgcn_wmma_f32_16x16x4_f32(
          false, ga, false, bgf, (short)0, accg[nt], false, false);
      acci[nt] = __builtin_amdgcn_wmma_f32_16x16x4_f32(
          false, ia, false, bif, (short)0, acci[nt], false, false);
    }
  }

  float bgl[4], bil[4];
#pragma unroll
  for (int nt = 0; nt < 4; nt++) {
    bgl[nt] = sbg[nt * 16 + l15];
    bil[nt] = sbi[nt * 16 + l15];
  }

  // C/D layout: VGPR j -> row m0+j (lanes 0-15) / m0+j+8 (lanes 16-31), col = nt*16+l15
#pragma unroll
  for (int j = 0; j < 8; j++) {
    int row = m0 + j + 8 * kh;
    float v[4];
    float sq = 0.0f;
#pragma unroll
    for (int nt = 0; nt < 4; nt++) {
      float g = accg[nt][j] + bgl[nt];
      float s = acci[nt][j] + bil[nt];
      g = (g > 0.0f) ? g : LEAKY * g;
      s = (s > 0.0f) ? s : LEAKY * s;
      v[nt] = g + s;
      sq += v[nt] * v[nt];
    }
    // row sum lives across the 16 lanes of this lane-half: 4-step butterfly
#pragma unroll
    for (int m = 1; m < 16; m <<= 1) sq += __shfl_xor(sq, m, 32);
    float inv = rsqrtf(fmaxf(sq, L2EPS));
    if (row < N) {
      float* hn = hnext + (size_t)row * 64 + l15;
      float* op = out + (size_t)row * outStride + colOff + l15;
#pragma unroll
      for (int nt = 0; nt < 4; nt++) {
        hn[nt * 16] = v[nt];        // raw h for next layer
        op[nt * 16] = v[nt] * inv;  // l2-normalized output
      }
    }
  }
}

// ---------------- launcher ----------------

extern "C" void kernel_launch(void* const* d_in, const int* in_sizes, int n_in,
                              void* d_out, int out_size, void* d_ws, size_t ws_size,
                              hipStream_t stream) {
  const float* x  = (const float*)d_in[0];
  const int*   ei = (const int*)d_in[1];
  const float* Wg = (const float*)d_in[2];
  const float* bg = (const float*)d_in[3];
  const float* Wi = (const float*)d_in[4];
  const float* bi = (const float*)d_in[5];

  const int d = 64;
  const int N = in_sizes[0] / d;
  const int E = in_sizes[1] / 2;
  const int K = in_sizes[2] / (d * d);
  const int outStride = (K + 1) * d;

  float* ws    = (float*)d_ws;
  float* dinv  = ws;                         // N   (deg, then rsqrt in place)
  float* wvec  = dinv + N;                   // E
  float* haBuf = wvec + E;                   // N*d
  float* hA    = haBuf + (size_t)N * d;      // N*d
  float* hB    = hA + (size_t)N * d;         // N*d
  float* out   = (float*)d_out;

  fill_zero_f32<<<(N + 255) / 256, 256, 0, stream>>>(dinv, N);
  deg_kernel<<<(E + 255) / 256, 256, 0, stream>>>(ei, dinv, E);
  dinv_kernel<<<(N + 255) / 256, 256, 0, stream>>>(dinv, N);
  edgew_kernel<<<(E + 255) / 256, 256, 0, stream>>>(ei, dinv, wvec, E);
  copy_x_kernel<<<(N * 16 + 255) / 256, 256, 0, stream>>>(x, out, N, outStride);

  const float* h = x;
  float* nxt = hA;
  int tiles = (N + 15) / 16;
  int lblocks = (tiles + 7) / 8;
  long long aggT = (long long)E * 32;
  int ablocks = (int)((aggT + 255) / 256);

  for (int layer = 0; layer < K; layer++) {
    fill_zero_f32<<<(N * d + 255) / 256, 256, 0, stream>>>(haBuf, N * d);
    agg_kernel<<<ablocks, 256, 0, stream>>>(ei, wvec, h, haBuf, E);
    layer_kernel<<<lblocks, 256, 0, stream>>>(
        h, haBuf,
        Wg + (size_t)layer * d * d, bg + (size_t)layer * d,
        Wi + (size_t)layer * d * d, bi + (size_t)layer * d,
        nxt, out, N, outStride, (layer + 1) * d);
    h = nxt;
    nxt = (nxt == hA) ? hB : hA;
  }
}